// BinaryBatchNorm_22634477650243
// MI455X (gfx1250) — compile-verified
//
#include <hip/hip_runtime.h>
#include <hip/hip_bf16.h>

// ---------------------------------------------------------------------------
// BinaryBatchNorm forward for MI455X (gfx1250), compile-only tuned.
// x: (64, 64, 112, 112) f32, NCHW.  Memory-bound: ~822MB moved => ~35us @ 23.3TB/s.
// Uses gfx1250 async global->LDS double-buffered streaming for the reductions.
// ---------------------------------------------------------------------------

#define C_CH      64
#define N_B       64
#define HW        12544                 // 112*112
#define VEC_SLAB  (HW / 4)              // 3136 float4 per (n,c) slab
#define THREADS   256
#define RED_ITERS 12                    // 12*256 = 3072 vec4 via async pipeline
#define TAIL      (VEC_SLAB - RED_ITERS * THREADS)   // 64 vec4 tail
#define NHW_INV   (1.0f / 802816.0f)
#define MOM       0.125f
#define EPSV      1e-5f

typedef float f32x4 __attribute__((ext_vector_type(4)));
typedef int   v4i   __attribute__((vector_size(16)));          // matches builtin param
typedef __attribute__((address_space(1))) v4i gv4i;            // global v4i
typedef __attribute__((address_space(3))) v4i lv4i;            // LDS v4i

#if defined(__has_builtin)
#  if __has_builtin(__builtin_amdgcn_global_load_async_to_lds_b128)
#    define USE_ASYNC_BUILTIN 1
#  endif
#  if __has_builtin(__builtin_amdgcn_s_wait_asynccnt)
#    define USE_WAIT_BUILTIN 1
#  endif
#endif

__device__ __forceinline__ void async_copy16(const f32x4* g, f32x4* l) {
#if defined(USE_ASYNC_BUILTIN)
  __builtin_amdgcn_global_load_async_to_lds_b128(
      (gv4i*)(unsigned long long)g,
      (lv4i*)(unsigned)(unsigned long long)l,
      0, 0);
#else
  unsigned lds_addr = (unsigned)(unsigned long long)l;   // flat low 32b == LDS byte addr
  asm volatile("global_load_async_to_lds_b128 %0, %1, off"
               :: "v"(lds_addr), "v"(g) : "memory");
#endif
}

template <int N>
__device__ __forceinline__ void wait_async() {
#if defined(USE_WAIT_BUILTIN)
  __builtin_amdgcn_s_wait_asynccnt((unsigned short)N);
#else
  asm volatile("s_wait_asynccnt %0" :: "i"(N));
#endif
  asm volatile("" ::: "memory");   // keep LDS reads after the wait
}

// approx_pow2(x) = sign(x) * 2^round(log2(|x|)).  Bit-exact for normals:
// round the exponent up iff mantissa > mantissa(sqrt(2)) = 0x3504F3.
__device__ __forceinline__ float ap2(float x) {
  unsigned u = __float_as_uint(x);
  unsigned s = u & 0x80000000u;
  unsigned a = u & 0x7fffffffu;
  if (a == 0u) return __uint_as_float(s);          // +-0 -> +-0
  unsigned e = a >> 23;
  if (e == 255u) return x;                         // inf / NaN propagate
  if (e == 0u) {                                   // denormal (rare) -> libm path
    float v = exp2f(rintf(log2f(__uint_as_float(a))));
    return __uint_as_float(__float_as_uint(v) | s);
  }
  unsigned m = a & 0x007fffffu;
  e += (m > 0x3504F3u) ? 1u : 0u;
  if (e >= 255u) return __uint_as_float(s | 0x7f800000u);
  return __uint_as_float(s | (e << 23));
}

template <int MODE>
__device__ __forceinline__ float accum4(f32x4 d, float mean) {
  if (MODE == 0) {
    return (d.x + d.y) + (d.z + d.w);
  } else {
    float t0 = d.x - mean, t1 = d.y - mean, t2 = d.z - mean, t3 = d.w - mean;
    return (t0 * ap2(t0) + t1 * ap2(t1)) + (t2 * ap2(t2) + t3 * ap2(t3));
  }
}

// One block per (c,n) slab: streams 12544 contiguous floats of channel c,
// double-buffered through LDS with async-tensor-path loads.
// MODE 0: partial[c*64+n] = sum(x).  MODE 1: partial = sum(ctr*ap2(ctr)).
template <int MODE>
__global__ void __launch_bounds__(THREADS)
reduce_kernel(const float* __restrict__ x, const float* __restrict__ chmean,
              float* __restrict__ partial) {
  __shared__ f32x4 buf[2][THREADS];
  __shared__ float red[THREADS];

  const int blk = blockIdx.x;
  const int c = blk >> 6;
  const int n = blk & 63;
  const int tid = threadIdx.x;
  const f32x4* v = (const f32x4*)(x + (size_t)(n * C_CH + c) * HW);
  const float mean = (MODE == 1) ? chmean[c] : 0.0f;

  float acc = 0.0f;
  async_copy16(v + tid, &buf[0][tid]);
#pragma unroll 1
  for (int it = 0; it < RED_ITERS; ++it) {
    if (it + 1 < RED_ITERS) {
      async_copy16(v + (it + 1) * THREADS + tid, &buf[(it + 1) & 1][tid]);
      wait_async<1>();          // older (current) buffer complete, next in flight
    } else {
      wait_async<0>();
    }
    acc += accum4<MODE>(buf[it & 1][tid], mean);
  }
  if (tid < TAIL) {             // 64 vec4 tail, plain b128 loads
    acc += accum4<MODE>(v[RED_ITERS * THREADS + tid], mean);
  }

  red[tid] = acc;
  __syncthreads();
#pragma unroll
  for (int s = THREADS / 2; s > 0; s >>= 1) {
    if (tid < s) red[tid] += red[tid + s];
    __syncthreads();
  }
  if (tid == 0) partial[blk] = red[0];
}

// 64 threads: fold 64 partials per channel deterministically -> blended mean.
__global__ void finalize_mean_kernel(const float* __restrict__ partial,
                                     const float* __restrict__ running_mean,
                                     float* __restrict__ chmean) {
  int c = threadIdx.x;
  float s = 0.0f;
#pragma unroll 1
  for (int n = 0; n < N_B; ++n) s += partial[c * N_B + n];
  chmean[c] = (1.0f - MOM) * running_mean[c] + MOM * (s * NHW_INV);
}

// 64 threads: blended var -> inv_std_q = ap2(1/sqrt(var+eps)); wq = ap2(weight).
__global__ void finalize_var_kernel(const float* __restrict__ partial,
                                    const float* __restrict__ running_var,
                                    const float* __restrict__ weight,
                                    float* __restrict__ invstd,
                                    float* __restrict__ wq) {
  int c = threadIdx.x;
  float s = 0.0f;
#pragma unroll 1
  for (int n = 0; n < N_B; ++n) s += partial[c * N_B + n];
  float var = (1.0f - MOM) * running_var[c] + MOM * (s * NHW_INV);
  invstd[c] = ap2(1.0f / sqrtf(var + EPSV));
  wq[c] = ap2(weight[c]);
}

// Elementwise pass: y = ap2(w)*ap2(ap2(x-mean)*inv_std_q) + bias.
// One float4 per lane; non-temporal hints stream x in and y out.
__global__ void __launch_bounds__(THREADS)
output_kernel(const float* __restrict__ x, const float* __restrict__ chmean,
              const float* __restrict__ invstd, const float* __restrict__ wq,
              const float* __restrict__ bias, float* __restrict__ y) {
  const int i = blockIdx.x * THREADS + threadIdx.x;   // vec4 index
  const int c = (i / VEC_SLAB) & (C_CH - 1);          // slab = n*64 + c
  const float m = chmean[c];
  const float is = invstd[c];
  const float w = wq[c];
  const float b = bias[c];

  const f32x4* xv = (const f32x4*)x;
  f32x4* yv = (f32x4*)y;
  f32x4 d = __builtin_nontemporal_load(xv + i);
  f32x4 r;
  r.x = w * ap2(ap2(d.x - m) * is) + b;
  r.y = w * ap2(ap2(d.y - m) * is) + b;
  r.z = w * ap2(ap2(d.z - m) * is) + b;
  r.w = w * ap2(ap2(d.w - m) * is) + b;
  __builtin_nontemporal_store(r, yv + i);
}

extern "C" void kernel_launch(void* const* d_in, const int* in_sizes, int n_in,
                              void* d_out, int out_size, void* d_ws, size_t ws_size,
                              hipStream_t stream) {
  const float* x  = (const float*)d_in[0];
  const float* w  = (const float*)d_in[1];
  const float* b  = (const float*)d_in[2];
  const float* rm = (const float*)d_in[3];
  const float* rv = (const float*)d_in[4];
  float* y = (float*)d_out;

  float* ws      = (float*)d_ws;
  float* partial = ws;                 // [64 channels][64 slabs]
  float* chmean  = ws + 4096;          // [64]
  float* invstd  = ws + 4096 + 64;     // [64]
  float* wq      = ws + 4096 + 128;    // [64]

  reduce_kernel<0><<<C_CH * N_B, THREADS, 0, stream>>>(x, chmean, partial);
  finalize_mean_kernel<<<1, C_CH, 0, stream>>>(partial, rm, chmean);
  reduce_kernel<1><<<C_CH * N_B, THREADS, 0, stream>>>(x, chmean, partial);
  finalize_var_kernel<<<1, C_CH, 0, stream>>>(partial, rv, w, invstd, wq);

  const int nvec = (N_B * C_CH * HW) / 4;   // 12,845,056
  output_kernel<<<nvec / THREADS, THREADS, 0, stream>>>(x, chmean, invstd, wq, b, y);
}